// ProjectionLoss_59390807769296
// MI455X (gfx1250) — compile-verified
//
#include <hip/hip_runtime.h>
#include <math.h>

// ---------------------------------------------------------------------------
// ProjectionLoss (Sinkhorn OT + alignment MSE) for MI455X / gfx1250
//   N=4096 descriptors, M=8192 embeddings, d=512, reg=0.1, 200 iterations
// Strategy:
//   * C (4096x8192 f32, 128MB) materialized once in d_ws -> fits 192MB L2;
//     all 400 Sinkhorn sweeps then run at L2 bandwidth, not HBM.
//   * GEMM uses V_WMMA_F32_16X16X4_F32 (full f32 = reference precision; GEMM
//     is <5% of runtime so reduced precision buys nothing). Each wave computes
//     a 16x64 strip: A fragment is register-reused across 4 WMMAs
//     (1.25 loads per WMMA instead of 2).
//   * Sinkhorn sweeps: single-read online logsumexp in base-2 (native
//     v_exp_f32/v_log_f32), quad-batched so the running-max rescale costs one
//     TRANS per 4 elements (1.25 exp/elem) and the loop-carried chain is 4x
//     shorter.
//   * Deterministic tree reductions only (graph-replayed, must be stable).
// Workspace: 128MB (C) + ~1.2MB scratch.
// ---------------------------------------------------------------------------

#define Nrow 4096
#define Mcol 8192
#define Dk   512

__device__ __constant__ const float REGc  = 0.1f;
#define LN2f   0.69314718055994530942f
#define K2f    14.426950408889634f        /* 1/(reg*ln2), reg=0.1 */
#define LOGAf  (-8.31776616671934f)       /* -ln(4096) */

typedef __attribute__((ext_vector_type(2))) float v2f;
typedef __attribute__((ext_vector_type(8))) float v8f;

// ---------------- helpers ----------------
__device__ __forceinline__ void lse_merge(float& m, float& s, float m2, float s2) {
    float M = fmaxf(m, m2);
    s = s * exp2f(m - M) + s2 * exp2f(m2 - M);
    m = M;
}

// Quad-batched online LSE step: one rescale TRANS per 4 elements.
__device__ __forceinline__ void lse_quad(float& m, float& s,
                                         float v0, float v1, float v2, float v3) {
    float mq = fmaxf(fmaxf(v0, v1), fmaxf(v2, v3));
    float nm = fmaxf(m, mq);
    s = s * exp2f(m - nm)
      + (exp2f(v0 - nm) + exp2f(v1 - nm))
      + (exp2f(v2 - nm) + exp2f(v3 - nm));
    m = nm;
}

// ---------------- row squared norms: one wave per row ----------------
__global__ void k_row_norms(const float* __restrict__ X, float* __restrict__ out, int nrows) {
    int tid = threadIdx.x, w = tid >> 5, lane = tid & 31;
    int i = blockIdx.x * 8 + w;
    if (i >= nrows) return;
    const float* x = X + (size_t)i * Dk;
    float s = 0.f;
    for (int k = lane; k < Dk; k += 32) { float v = x[k]; s += v * v; }
    for (int off = 16; off; off >>= 1) s += __shfl_down(s, off, 32);
    if (lane == 0) out[i] = s;
}

// ---------------- sum of tgt_probs (single block, deterministic) ----------------
__global__ void k_sumtp(const float* __restrict__ tp, float* __restrict__ out) {
    __shared__ float sd[256];
    int tid = threadIdx.x;
    float s = 0.f;
    for (int j = tid; j < Mcol; j += 256) s += tp[j];
    sd[tid] = s; __syncthreads();
    for (int off = 128; off; off >>= 1) { if (tid < off) sd[tid] += sd[tid + off]; __syncthreads(); }
    if (tid == 0) out[0] = sd[0];
}

// ---------------- GEMM + sqrt-dist epilogue via f32 WMMA ----------------
// Block = 256 thr = 8 waves stacked along M -> block tile 128(M) x 64(N).
// Each wave computes a 16x64 strip: A fragment loaded once per k-step and
// register-reused across 4 WMMAs (4 B fragments at +16 column steps).
// A operand (16x4): lanes 0-15 hold {K=k0,k0+1} of row M=lane; lanes 16-31
// hold {K=k0+2,k0+3} of row M=lane-16.  B (4x16) symmetric over W rows
// (dot_ij = sum_k X[i,k]*W[j,k], so B columns come from W rows).
__global__ void k_gemm(const float* __restrict__ X, const float* __restrict__ W,
                       const float* __restrict__ xx, const float* __restrict__ yy,
                       float* __restrict__ C) {
    int tid  = threadIdx.x;
    int w    = tid >> 5, lane = tid & 31;
    int half = lane >> 4, lm = lane & 15;
    int m0 = blockIdx.x * 128 + w * 16;
    int n0 = blockIdx.y * 64;

    const float* pA = X + (size_t)(m0 + lm) * Dk + 2 * half;
    const float* pB = W + (size_t)(n0 + lm) * Dk + 2 * half;   // +t*16 rows => +t*16*Dk

    v8f acc0 = {}, acc1 = {}, acc2 = {}, acc3 = {};
    for (int k0 = 0; k0 < Dk; k0 += 4) {
        v2f a  = *(const v2f*)(pA + k0);
        v2f b0 = *(const v2f*)(pB + k0);
        v2f b1 = *(const v2f*)(pB + k0 + 16 * Dk);
        v2f b2 = *(const v2f*)(pB + k0 + 32 * Dk);
        v2f b3 = *(const v2f*)(pB + k0 + 48 * Dk);
        acc0 = __builtin_amdgcn_wmma_f32_16x16x4_f32(false, a, false, b0, (short)0, acc0, false, false);
        acc1 = __builtin_amdgcn_wmma_f32_16x16x4_f32(false, a, false, b1, (short)0, acc1, false, false);
        acc2 = __builtin_amdgcn_wmma_f32_16x16x4_f32(false, a, false, b2, (short)0, acc2, false, false);
        acc3 = __builtin_amdgcn_wmma_f32_16x16x4_f32(false, a, false, b3, (short)0, acc3, false, false);
    }

    // C/D layout: VGPR r, lanes 0-15 -> M = r, lanes 16-31 -> M = 8 + r.
    for (int t = 0; t < 4; ++t) {
        v8f acc = (t == 0) ? acc0 : (t == 1) ? acc1 : (t == 2) ? acc2 : acc3;
        int j = n0 + 16 * t + lm;
        float yyj = yy[j];
        for (int r = 0; r < 8; ++r) {
            int i = m0 + 8 * half + r;
            float sq = xx[i] + yyj - 2.0f * acc[r];
            C[(size_t)i * Mcol + j] = sqrtf(fmaxf(sq, 1e-12f));
        }
    }
}

// ---------------- zero-init g ----------------
__global__ void k_zero(float* __restrict__ g) {
    g[blockIdx.x * 256 + threadIdx.x] = 0.f;
}

// ---------------- f update: one block per row, quad-batched online LSE ----------------
__global__ void k_f(const float* __restrict__ C, const float* __restrict__ g,
                    float* __restrict__ f) {
    __shared__ float sm[256], ss[256];
    int i = blockIdx.x, tid = threadIdx.x;
    const float* row = C + (size_t)i * Mcol;
    float m = -INFINITY, s = 0.f;
    for (int j = tid; j < Mcol; j += 1024) {          // 8 quads of strided elements
        float v0 = (g[j       ] - row[j       ]) * K2f;
        float v1 = (g[j +  256] - row[j +  256]) * K2f;
        float v2 = (g[j +  512] - row[j +  512]) * K2f;
        float v3 = (g[j +  768] - row[j +  768]) * K2f;
        lse_quad(m, s, v0, v1, v2, v3);
    }
    sm[tid] = m; ss[tid] = s; __syncthreads();
    for (int off = 128; off; off >>= 1) {
        if (tid < off) { float mm = sm[tid], sv = ss[tid];
                         lse_merge(mm, sv, sm[tid + off], ss[tid + off]);
                         sm[tid] = mm; ss[tid] = sv; }
        __syncthreads();
    }
    if (tid == 0) f[i] = REGc * (LOGAf - LN2f * (sm[0] + log2f(ss[0])));
}

// ---------------- g update, phase 1: per-column partial LSE over a 256-row slab ----
__global__ void k_gp(const float* __restrict__ C, const float* __restrict__ f,
                     float* __restrict__ pm, float* __restrict__ ps) {
    __shared__ float sf[256];
    int tid  = threadIdx.x;
    int j    = blockIdx.x * 256 + tid;
    int rbeg = blockIdx.y * 256;
    sf[tid] = f[rbeg + tid];
    __syncthreads();
    float m = -INFINITY, s = 0.f;
    const float* p = C + (size_t)rbeg * Mcol + j;
    for (int ii = 0; ii < 256; ii += 4) {             // 64 quads of consecutive rows
        float v0 = (sf[ii    ] - p[(size_t)(ii    ) * Mcol]) * K2f;
        float v1 = (sf[ii + 1] - p[(size_t)(ii + 1) * Mcol]) * K2f;
        float v2 = (sf[ii + 2] - p[(size_t)(ii + 2) * Mcol]) * K2f;
        float v3 = (sf[ii + 3] - p[(size_t)(ii + 3) * Mcol]) * K2f;
        lse_quad(m, s, v0, v1, v2, v3);
    }
    pm[(size_t)blockIdx.y * Mcol + j] = m;
    ps[(size_t)blockIdx.y * Mcol + j] = s;
}

// ---------------- g update, phase 2: merge 16 slabs, apply logb ----------------
__global__ void k_gc(const float* __restrict__ pm, const float* __restrict__ ps,
                     const float* __restrict__ tp, const float* __restrict__ sumtp,
                     float* __restrict__ g) {
    int j = blockIdx.x * 256 + threadIdx.x;
    float m = pm[j], s = ps[j];
    for (int p = 1; p < 16; ++p)
        lse_merge(m, s, pm[(size_t)p * Mcol + j], ps[(size_t)p * Mcol + j]);
    float logb = logf(tp[j]) - logf(sumtp[0]);
    g[j] = REGc * (logb - LN2f * (m + log2f(s)));
}

// ---------------- final <P, C>: one block per row ----------------
__global__ void k_loss(const float* __restrict__ C, const float* __restrict__ f,
                       const float* __restrict__ g, float* __restrict__ rowloss) {
    __shared__ float sd[256];
    int i = blockIdx.x, tid = threadIdx.x;
    float fi = f[i];
    const float* row = C + (size_t)i * Mcol;
    float acc = 0.f;
    for (int j = tid; j < Mcol; j += 256) {
        float c = row[j];
        acc += exp2f((fi + g[j] - c) * K2f) * c;
    }
    sd[tid] = acc; __syncthreads();
    for (int off = 128; off; off >>= 1) { if (tid < off) sd[tid] += sd[tid + off]; __syncthreads(); }
    if (tid == 0) rowloss[i] = sd[0];
}

// ---------------- alignment MSE partials: one wave per row ----------------
__global__ void k_dist(const float* __restrict__ X, const float* __restrict__ W,
                       const int* __restrict__ aligned,
                       float* __restrict__ dsq, float* __restrict__ dcnt) {
    __shared__ float wsq[8], wcnt[8];
    int tid = threadIdx.x, w = tid >> 5, lane = tid & 31;
    int i = blockIdx.x * 8 + w;
    int a = aligned[i];
    float maskf = (a != -1) ? 1.f : 0.f;
    int idx = (a < 0) ? 0 : a;
    const float* x = X + (size_t)i * Dk;
    const float* y = W + (size_t)idx * Dk;
    float s = 0.f;
    for (int k = lane; k < Dk; k += 32) { float d = x[k] - y[k]; s += d * d; }
    for (int off = 16; off; off >>= 1) s += __shfl_down(s, off, 32);
    if (lane == 0) { wsq[w] = s * maskf; wcnt[w] = maskf; }
    __syncthreads();
    if (tid == 0) {
        float a0 = 0.f, c0 = 0.f;
        for (int q = 0; q < 8; ++q) { a0 += wsq[q]; c0 += wcnt[q]; }
        dsq[blockIdx.x] = a0; dcnt[blockIdx.x] = c0;
    }
}

// ---------------- final deterministic combine ----------------
__global__ void k_final(const float* __restrict__ rowloss, const float* __restrict__ dsq,
                        const float* __restrict__ dcnt, float* __restrict__ out) {
    __shared__ float s0[256], s1[256], s2[256];
    int tid = threadIdx.x;
    float ot = 0.f;
    for (int i = tid; i < Nrow; i += 256) ot += rowloss[i];
    float sq = 0.f, cnt = 0.f;
    for (int i = tid; i < 512; i += 256) { sq += dsq[i]; cnt += dcnt[i]; }
    s0[tid] = ot; s1[tid] = sq; s2[tid] = cnt; __syncthreads();
    for (int off = 128; off; off >>= 1) {
        if (tid < off) { s0[tid] += s0[tid + off]; s1[tid] += s1[tid + off]; s2[tid] += s2[tid + off]; }
        __syncthreads();
    }
    if (tid == 0) {
        float dist = (s2[0] > 0.f) ? s1[0] / (s2[0] * (float)Dk) : 0.f;
        out[0] = s0[0] + dist;
    }
}

// ---------------------------------------------------------------------------
extern "C" void kernel_launch(void* const* d_in, const int* in_sizes, int n_in,
                              void* d_out, int out_size, void* d_ws, size_t ws_size,
                              hipStream_t stream) {
    const float* X  = (const float*)d_in[0];   // descriptors      (4096*512)
    const float* W  = (const float*)d_in[1];   // word_embeddings  (8192*512)
    const int*   al = (const int*)  d_in[2];   // aligned          (4096)
    const float* tp = (const float*)d_in[3];   // tgt_probs        (8192)
    float* out = (float*)d_out;
    float* ws  = (float*)d_ws;

    const size_t NC = (size_t)Nrow * Mcol;     // 32M floats = 128 MB
    float* C       = ws;
    float* xx      = C + NC;                   // 4096
    float* yy      = xx + Nrow;                // 8192
    float* f       = yy + Mcol;                // 4096
    float* g       = f + Nrow;                 // 8192
    float* pm      = g + Mcol;                 // 16*8192
    float* ps      = pm + 16 * Mcol;           // 16*8192
    float* rowloss = ps + 16 * Mcol;           // 4096
    float* dsq     = rowloss + Nrow;           // 512
    float* dcnt    = dsq + 512;                // 512
    float* sumtp   = dcnt + 512;               // 1

    k_row_norms<<<Nrow / 8, 256, 0, stream>>>(X, xx, Nrow);
    k_row_norms<<<Mcol / 8, 256, 0, stream>>>(W, yy, Mcol);
    k_sumtp<<<1, 256, 0, stream>>>(tp, sumtp);

    k_gemm<<<dim3(Nrow / 128, Mcol / 64), 256, 0, stream>>>(X, W, xx, yy, C);

    k_zero<<<Mcol / 256, 256, 0, stream>>>(g);
    for (int it = 0; it < 200; ++it) {
        k_f <<<Nrow, 256, 0, stream>>>(C, g, f);
        k_gp<<<dim3(Mcol / 256, Nrow / 256), 256, 0, stream>>>(C, f, pm, ps);
        k_gc<<<Mcol / 256, 256, 0, stream>>>(pm, ps, tp, sumtp, g);
    }

    k_loss <<<Nrow, 256, 0, stream>>>(C, f, g, rowloss);
    k_dist <<<Nrow / 8, 256, 0, stream>>>(X, W, al, dsq, dcnt);
    k_final<<<1, 256, 0, stream>>>(rowloss, dsq, dcnt, out);
}